// OccupancyAwareLifting_75170517614723
// MI455X (gfx1250) — compile-verified
//
#include <hip/hip_runtime.h>
#include <hip/hip_bf16.h>
#include <math.h>
#include <stdint.h>

#define DEV static __device__ __forceinline__

namespace {
constexpr int B_ = 2, K_ = 12, H_ = 120, W_ = 160, D_ = 128, NB_ = 100;
constexpr int P_ = H_ * W_;                               // 19200 pixels / batch
constexpr long long VOX_ = (long long)B_ * D_ * D_ * D_;  // 4,194,304 voxels
constexpr float DMAX_ = 6.0f;
}

typedef float oal_f4 __attribute__((ext_vector_type(4)));  // builtin-friendly float4

// ---------------------------------------------------------------------------
// CDNA5 async global->LDS path (ASYNCcnt), guarded so compile never breaks.
// Builtin signature (probe-confirmed): (as1 int* src, as3 int* dst, imm, imm)
// ---------------------------------------------------------------------------
#if defined(__HIP_DEVICE_COMPILE__) && __has_builtin(__builtin_amdgcn_global_load_async_to_lds_b32)
#define USE_ASYNC_LDS 1
typedef __attribute__((address_space(1))) int gas_int;
typedef __attribute__((address_space(3))) int las_int;
DEV void oal_async_copy_f32(const float* gsrc, float* lds_dst) {
  __builtin_amdgcn_global_load_async_to_lds_b32(
      (gas_int*)(unsigned long long)(uintptr_t)gsrc,
      (las_int*)(unsigned int)(uintptr_t)lds_dst, 0, 0);
}
DEV void oal_wait_async_le1() {
#if __has_builtin(__builtin_amdgcn_s_wait_asynccnt)
  __builtin_amdgcn_s_wait_asynccnt(1);
#else
  asm volatile("s_wait_asynccnt 0x1" ::: "memory");
#endif
}
DEV void oal_wait_async_le0() {
#if __has_builtin(__builtin_amdgcn_s_wait_asynccnt)
  __builtin_amdgcn_s_wait_asynccnt(0);
#else
  asm volatile("s_wait_asynccnt 0x0" ::: "memory");
#endif
}
#else
#define USE_ASYNC_LDS 0
#endif

// ---------------------------------------------------------------------------
// 4) fused features writer: [B, 15, 100, H, W]; 230 MB of coalesced NT stores.
//    Occupancy stream staged through LDS with async global->LDS double buffer.
//    (Placed first in the TU so the disasm snippet shows the async path.)
// ---------------------------------------------------------------------------
__global__ __launch_bounds__(256) void oal_features(
    const float* __restrict__ feats_sm, const int* __restrict__ dindex,
    const float* __restrict__ stuff_x, const float* __restrict__ stuff_y,
    const float* __restrict__ occ, float* __restrict__ feat_out) {
  int p = blockIdx.x * 256 + threadIdx.x;   // grid exactly covers P_
  int b = blockIdx.y;
  if (p >= P_) return;
  int h = p / W_, w = p % W_;
  float f[K_];
  const float* fs = feats_sm + (size_t)b * K_ * P_ + p;
#pragma unroll
  for (int k = 0; k < K_; ++k) f[k] = fs[(size_t)k * P_];
  int di = dindex[b * P_ + p];
  float dmf = fminf(stuff_x[b * W_ + w], stuff_y[b * H_ + h]);
  int dmb = (int)(dmf / DMAX_ * 100.0f);
  const float* op = occ + (size_t)b * NB_ * P_ + p;
  float* ob = feat_out + (size_t)b * 15 * NB_ * P_ + p;

#if USE_ASYNC_LDS
  __shared__ float stage[2][256];
  oal_async_copy_f32(op, &stage[0][threadIdx.x]);   // prefetch bin 0
#endif
  for (int bin = 0; bin < NB_; ++bin) {
    float o;
#if USE_ASYNC_LDS
    if (bin + 1 < NB_) {
      oal_async_copy_f32(op + (size_t)(bin + 1) * P_, &stage[(bin + 1) & 1][threadIdx.x]);
      oal_wait_async_le1();
    } else {
      oal_wait_async_le0();
    }
    o = stage[bin & 1][threadIdx.x];
#else
    __builtin_prefetch(op + (size_t)(bin + 4) * P_, 0, 1);
    o = op[(size_t)bin * P_];
#endif
    bool dwk = (bin > di - 3) && (bin < dmb + 5);
    float dw = dwk ? (1.0f / (1.0f + __expf(-o))) : 0.0f;
    int dd = bin - di;
    float sgn = (dd > 0) ? 1.0f : ((dd < 0) ? -1.0f : 0.0f);
    float ab = fabsf((float)dd) * (DMAX_ / 100.0f);
    // one-shot 230MB output stream: non-temporal stores keep L2 for reused data
#pragma unroll
    for (int k = 0; k < K_; ++k)
      __builtin_nontemporal_store(f[k], &ob[(size_t)(k * NB_ + bin) * P_]);
    __builtin_nontemporal_store(dw,  &ob[(size_t)(12 * NB_ + bin) * P_]);
    __builtin_nontemporal_store(sgn, &ob[(size_t)(13 * NB_ + bin) * P_]);
    __builtin_nontemporal_store(ab,  &ob[(size_t)(14 * NB_ + bin) * P_]);
  }
}

// ---------------------------------------------------------------------------
// 0) zero tiny reduction scratch
// ---------------------------------------------------------------------------
__global__ void oal_zero(unsigned int* xmax, unsigned int* ymax, unsigned int* cnt) {
  int t = blockIdx.x * blockDim.x + threadIdx.x;
  if (t < B_ * W_) xmax[t] = 0u;
  if (t < B_ * H_) ymax[t] = 0u;
  if (t < B_) cnt[t] = 0u;
}

// ---------------------------------------------------------------------------
// 1) per-pixel: softmax over K, argmax>=10 flag, depth clamp + bin index
// ---------------------------------------------------------------------------
__global__ void oal_pixel_prep(const float* __restrict__ sem2d,
                               const float* __restrict__ depth,
                               float* __restrict__ feats_sm,
                               unsigned char* __restrict__ ge10,
                               int* __restrict__ dindex) {
  int t = blockIdx.x * blockDim.x + threadIdx.x;
  if (t >= B_ * P_) return;
  int b = t / P_, p = t % P_;
  const float* s = sem2d + (size_t)b * K_ * P_ + p;
  float v[K_];
  float m = -INFINITY; int arg = 0;
#pragma unroll
  for (int k = 0; k < K_; ++k) { v[k] = s[(size_t)k * P_]; if (v[k] > m) { m = v[k]; arg = k; } }
  float sum = 0.f;
#pragma unroll
  for (int k = 0; k < K_; ++k) { v[k] = __expf(v[k] - m); sum += v[k]; }
  float inv = 1.0f / sum;
  float* fo = feats_sm + (size_t)b * K_ * P_ + p;
#pragma unroll
  for (int k = 0; k < K_; ++k) fo[(size_t)k * P_] = v[k] * inv;
  ge10[t] = (arg >= 10) ? 1 : 0;
  float d = fminf(depth[t], DMAX_);
  dindex[t] = (int)(d / DMAX_ * 100.0f);
}

// ---------------------------------------------------------------------------
// 2) 5x5 erosion (clipped window), stuff_depth, atomic col/row maxes (>=0 =>
//    float bit pattern is monotone as uint)
// ---------------------------------------------------------------------------
__global__ void oal_stuff(const unsigned char* __restrict__ ge10,
                          const float* __restrict__ depth,
                          float* __restrict__ stuff_depth,
                          unsigned int* __restrict__ xmax,
                          unsigned int* __restrict__ ymax) {
  int t = blockIdx.x * blockDim.x + threadIdx.x;
  if (t >= B_ * P_) return;
  int b = t / P_, p = t % P_, h = p / W_, w = p % W_;
  int s = 1;
  for (int dy = -2; dy <= 2; ++dy) {
    int hh = h + dy; if ((unsigned)hh >= (unsigned)H_) continue;
    for (int dx = -2; dx <= 2; ++dx) {
      int ww = w + dx; if ((unsigned)ww >= (unsigned)W_) continue;
      s &= (int)ge10[(size_t)b * P_ + hh * W_ + ww];
    }
  }
  float sd = s ? fminf(depth[t], DMAX_) : 0.0f;
  stuff_depth[t] = sd;
  unsigned int bits = __float_as_uint(sd);
  atomicMax(&xmax[b * W_ + w], bits);
  atomicMax(&ymax[b * H_ + h], bits);
}

// ---------------------------------------------------------------------------
// 3) per-batch sequential edge fill + column/row maxes + _find_none scans
// ---------------------------------------------------------------------------
DEV void oal_find_none(const float* a, int n, float* out) {
  float suf[W_];  // W_ >= H_
  float m = INFINITY;
  for (int i = n - 1; i >= 0; --i) { suf[i] = m; if (a[i] != 0.f) m = fminf(m, a[i]); }
  float run = INFINITY;
  for (int i = 0; i < n; ++i) {
    float l = isinf(run) ? 0.f : run;
    float r = isinf(suf[i]) ? 0.f : suf[i];
    float val = (a[i] == 0.f) ? fmaxf(l, r) : a[i];
    if (val != 0.f) run = fminf(run, val);
    out[i] = val;
  }
}

__global__ void oal_scan(const float* __restrict__ sd,
                         const unsigned int* __restrict__ xmaxb,
                         const unsigned int* __restrict__ ymaxb,
                         float* __restrict__ stuff_x, float* __restrict__ stuff_y) {
  int b = threadIdx.x;
  if (b >= B_) return;
  const float* SD = sd + (size_t)b * P_;
  // first-nonzero fill values of the 1D max vectors (0 if all-zero)
  float vx = 0.f, vxr = 0.f, vy = 0.f, vyr = 0.f;
  for (int w = 0; w < W_; ++w) { float u = __uint_as_float(xmaxb[b * W_ + w]); if (u != 0.f) { vx = u; break; } }
  for (int w = W_ - 1; w >= 0; --w) { float u = __uint_as_float(xmaxb[b * W_ + w]); if (u != 0.f) { vxr = u; break; } }
  for (int h = 0; h < H_; ++h) { float u = __uint_as_float(ymaxb[b * H_ + h]); if (u != 0.f) { vy = u; break; } }
  for (int h = H_ - 1; h >= 0; --h) { float u = __uint_as_float(ymaxb[b * H_ + h]); if (u != 0.f) { vyr = u; break; } }

  float colmax[W_], rowmax[H_];
  for (int w = 0; w < W_; ++w) {
    float m;
    if (w == 0 || w == W_ - 1) {       // whole column replaced by col-fill
      float fill = (w == 0) ? vx : vxr;
      m = 0.f;
      for (int h = 0; h < H_; ++h) { float v = SD[h * W_ + w]; if (v == 0.f) v = fill; m = fmaxf(m, v); }
    } else {                           // interior: edge rows may be filled
      float r0 = SD[w];                 if (r0 == 0.f) r0 = vy;
      float r1 = SD[(H_ - 1) * W_ + w]; if (r1 == 0.f) r1 = vyr;
      m = fmaxf(__uint_as_float(xmaxb[b * W_ + w]), fmaxf(r0, r1));
    }
    colmax[w] = m;
  }
  for (int h = 0; h < H_; ++h) {
    float m;
    if (h == 0 || h == H_ - 1) {
      float fill = (h == 0) ? vy : vyr;
      m = 0.f;
      for (int w = 0; w < W_; ++w) {
        float v = SD[h * W_ + w];
        float f = (w == 0) ? vx : ((w == W_ - 1) ? vxr : fill);  // corners use col fill
        if (v == 0.f) v = f;
        m = fmaxf(m, v);
      }
    } else {
      float r0 = SD[h * W_];            if (r0 == 0.f) r0 = vx;
      float r1 = SD[h * W_ + W_ - 1];   if (r1 == 0.f) r1 = vxr;
      m = fmaxf(__uint_as_float(ymaxb[b * H_ + h]), fmaxf(r0, r1));
    }
    rowmax[h] = m;
  }
  oal_find_none(colmax, W_, stuff_x + b * W_);
  oal_find_none(rowmax, H_, stuff_y + b * H_);
}

// ---------------------------------------------------------------------------
// 5) voxel gather: feat_kept = kept & dwk[b, dbin, y, x]
// ---------------------------------------------------------------------------
__global__ void oal_voxel(const unsigned char* __restrict__ kept,
                          const oal_f4* __restrict__ mapping,
                          const int* __restrict__ dindex,
                          const float* __restrict__ sx, const float* __restrict__ sy,
                          unsigned char* __restrict__ fk, float* __restrict__ fk_out) {
  long long t = (long long)blockIdx.x * blockDim.x + threadIdx.x;
  if (t >= VOX_) return;
  oal_f4 m = __builtin_nontemporal_load(&mapping[t]);  // 64MB one-shot stream
  int bi = (int)m.x, xi = (int)m.y, yi = (int)m.z;
  int dbin = (int)(m.w * 100.0f / DMAX_);
  int di = dindex[bi * P_ + yi * W_ + xi];
  int dmb = (int)(fminf(sx[bi * W_ + xi], sy[bi * H_ + yi]) / DMAX_ * 100.0f);
  int ok = (kept[t] != 0) && (dbin > di - 3) && (dbin < dmb + 5);
  fk[t] = (unsigned char)ok;
  __builtin_nontemporal_store(ok ? 1.0f : 0.0f, &fk_out[t]);
}

// ---------------------------------------------------------------------------
// separable 5-wide binary OR-pool along one axis (stride 1 / D / D*D)
// ---------------------------------------------------------------------------
__global__ void oal_pool_or(const unsigned char* __restrict__ in,
                            unsigned char* __restrict__ out, int stride,
                            float* __restrict__ fout) {
  long long t = (long long)blockIdx.x * blockDim.x + threadIdx.x;
  if (t >= VOX_) return;
  int c = (int)((t / stride) % D_);
  int acc = 0;
#pragma unroll
  for (int o = -2; o <= 2; ++o) {
    int cc = c + o;
    if ((unsigned)cc < (unsigned)D_) acc |= (int)in[t + (long long)o * stride];
  }
  if (fout) __builtin_nontemporal_store(acc ? 1.0f : 0.0f, &fout[t]);
  else out[t] = (unsigned char)(acc ? 1 : 0);
}

// 6) padding_kept = pool & kept; per-batch count via wave32 ballot
__global__ void oal_pk(const unsigned char* __restrict__ pool,
                       const unsigned char* __restrict__ kept,
                       unsigned char* __restrict__ pk, unsigned int* __restrict__ cnt) {
  long long t = (long long)blockIdx.x * blockDim.x + threadIdx.x;
  if (t >= VOX_) return;
  int v = (pool[t] != 0) & (kept[t] != 0);
  pk[t] = (unsigned char)v;
  unsigned long long bal = __ballot(v);
  if ((threadIdx.x & 31) == 0) {
    int b = (int)(t / ((long long)D_ * D_ * D_));
    atomicAdd(&cnt[b], (unsigned int)__popcll(bal));
  }
}

__global__ void oal_fix(unsigned char* pk, const unsigned int* cnt) {
  int b = threadIdx.x;
  if (b >= B_) return;
  if (cnt[b] == 0)
    pk[(long long)b * D_ * D_ * D_ + 127LL * D_ * D_ + 127 * D_ + 127] = 1;
}

// 7) per 2x2x2 block: padding_even at even corner, union = fk | padding_even
__global__ void oal_union(const unsigned char* __restrict__ pk,
                          const unsigned char* __restrict__ fk,
                          unsigned char* __restrict__ uni) {
  long long t = (long long)blockIdx.x * blockDim.x + threadIdx.x;
  const long long NBLK = (long long)B_ * 64 * 64 * 64;
  if (t >= NBLK) return;
  int b = (int)(t / (64 * 64 * 64));
  int r = (int)(t % (64 * 64 * 64));
  int i = r / (64 * 64), j = (r / 64) % 64, k = r % 64;
  long long base = ((long long)b * D_ + 2 * i) * D_ * D_ + (long long)(2 * j) * D_ + 2 * k;
  int any = 0;
#pragma unroll
  for (int di = 0; di < 2; ++di)
#pragma unroll
    for (int dj = 0; dj < 2; ++dj)
#pragma unroll
      for (int dk = 0; dk < 2; ++dk) {
        long long o = base + (long long)di * D_ * D_ + dj * D_ + dk;
        unsigned char f = fk[o];
        any |= ((pk[o] != 0) && !f);
        uni[o] = f;
      }
  if (any) uni[base] = 1;
}

// ---------------------------------------------------------------------------
extern "C" void kernel_launch(void* const* d_in, const int* in_sizes, int n_in,
                              void* d_out, int out_size, void* d_ws, size_t ws_size,
                              hipStream_t stream) {
  const float* sem   = (const float*)d_in[0];
  const float* depth = (const float*)d_in[1];
  const float* occ   = (const float*)d_in[2];
  const unsigned char* kept = (const unsigned char*)d_in[3];  // jax bool = 1 byte
  const oal_f4* mapping = (const oal_f4*)d_in[4];

  float* out_features = (float*)d_out;
  float* out_fk   = out_features + (size_t)B_ * 15 * NB_ * P_;
  float* out_mask = out_fk + (size_t)VOX_;

  char* wp = (char*)d_ws;
  auto carve = [&wp](size_t bytes) { void* p = (void*)wp; wp += (bytes + 255) & ~(size_t)255; return p; };
  float*         feats_sm = (float*)carve(sizeof(float) * (size_t)B_ * K_ * P_);
  int*           dindex   = (int*)carve(sizeof(int) * (size_t)B_ * P_);
  unsigned char* ge10     = (unsigned char*)carve((size_t)B_ * P_);
  float*         sdepth   = (float*)carve(sizeof(float) * (size_t)B_ * P_);
  unsigned int*  xmax     = (unsigned int*)carve(sizeof(int) * (size_t)B_ * W_);
  unsigned int*  ymax     = (unsigned int*)carve(sizeof(int) * (size_t)B_ * H_);
  float*         sx       = (float*)carve(sizeof(float) * (size_t)B_ * W_);
  float*         sy       = (float*)carve(sizeof(float) * (size_t)B_ * H_);
  unsigned int*  cnt      = (unsigned int*)carve(sizeof(int) * (size_t)B_);
  unsigned char* fk       = (unsigned char*)carve((size_t)VOX_);
  unsigned char* pa       = (unsigned char*)carve((size_t)VOX_);
  unsigned char* pb       = (unsigned char*)carve((size_t)VOX_);

  const int PIX_BLOCKS = (B_ * P_) / 256;       // 150, exact
  const int VOX_BLOCKS = (int)(VOX_ / 256);     // 16384, exact

  oal_zero<<<2, 256, 0, stream>>>(xmax, ymax, cnt);
  oal_pixel_prep<<<PIX_BLOCKS, 256, 0, stream>>>(sem, depth, feats_sm, ge10, dindex);
  oal_stuff<<<PIX_BLOCKS, 256, 0, stream>>>(ge10, depth, sdepth, xmax, ymax);
  oal_scan<<<1, 32, 0, stream>>>(sdepth, xmax, ymax, sx, sy);
  oal_features<<<dim3(P_ / 256, B_), 256, 0, stream>>>(feats_sm, dindex, sx, sy, occ, out_features);
  oal_voxel<<<VOX_BLOCKS, 256, 0, stream>>>(kept, mapping, dindex, sx, sy, fk, out_fk);
  // maxpool3d(feat_kept), separable
  oal_pool_or<<<VOX_BLOCKS, 256, 0, stream>>>(fk, pa, 1, nullptr);
  oal_pool_or<<<VOX_BLOCKS, 256, 0, stream>>>(pa, pb, D_, nullptr);
  oal_pool_or<<<VOX_BLOCKS, 256, 0, stream>>>(pb, pa, D_ * D_, nullptr);
  oal_pk<<<VOX_BLOCKS, 256, 0, stream>>>(pa, kept, pb, cnt);
  oal_fix<<<1, 32, 0, stream>>>(pb, cnt);
  oal_union<<<(int)(((long long)B_ * 64 * 64 * 64) / 256), 256, 0, stream>>>(pb, fk, pa);
  // maxpool3d(union) -> mask (float, written by last pass)
  oal_pool_or<<<VOX_BLOCKS, 256, 0, stream>>>(pa, pb, 1, nullptr);
  oal_pool_or<<<VOX_BLOCKS, 256, 0, stream>>>(pb, pa, D_, nullptr);
  oal_pool_or<<<VOX_BLOCKS, 256, 0, stream>>>(pa, nullptr, D_ * D_, out_mask);

  (void)in_sizes; (void)n_in; (void)out_size; (void)ws_size;
}